// GATModel_55946243998131
// MI455X (gfx1250) — compile-verified
//
#include <hip/hip_runtime.h>
#include <math.h>

#define F_IN 128
#define HC   128     // H*C
#define HEADS 2
#define CH   64
#define NC   16
#define NEG_SLOPE 0.2f

typedef __attribute__((ext_vector_type(2))) float v2f;
typedef __attribute__((ext_vector_type(8))) float v8f;

// ---------------- init scratch ----------------
__global__ void init_ws(float* __restrict__ agg, unsigned* __restrict__ m_u,
                        float* __restrict__ den, int Nn) {
    int i = blockIdx.x * blockDim.x + threadIdx.x;
    if (i < Nn * HC) agg[i] = 0.0f;
    if (i < Nn * HEADS) { m_u[i] = 0u; den[i] = 0.0f; }
}

// ---------------- h = x @ W via f32 WMMA 16x16x4 ----------------
// one wave computes a 16x16 tile; 8 waves/block cover 16 rows x 128 cols
__global__ __launch_bounds__(256) void gemm_wmma(const float* __restrict__ X,
                                                 const float* __restrict__ W,
                                                 float* __restrict__ Hout, int Nn) {
    const int lane = threadIdx.x & 31;
    const int wave = threadIdx.x >> 5;
    const int rowBase = blockIdx.x * 16;
    const int colBase = wave * 16;
    const int l    = lane & 15;
    const int half = lane >> 4;

    int arow = rowBase + l;
    if (arow >= Nn) arow = Nn - 1;            // clamp (Nn=50000 is divisible by 16 anyway)
    const float* xrow = X + (size_t)arow * F_IN;

    v8f acc = {};
    #pragma unroll
    for (int k = 0; k < F_IN; k += 4) {
        const int ka = k + 2 * half;
        v2f a, b;
        // A 16x4: lane(0-15)=rows, VGPR0/1 = K pair {0,1} (half=0) or {2,3} (half=1)
        a.x = xrow[ka];
        a.y = xrow[ka + 1];
        // B 4x16: lane(0-15)=cols, VGPR0/1 = K pair per half
        b.x = W[(size_t)ka * HC + colBase + l];
        b.y = W[(size_t)(ka + 1) * HC + colBase + l];
        acc = __builtin_amdgcn_wmma_f32_16x16x4_f32(
            /*neg_a=*/false, a, /*neg_b=*/false, b,
            /*c_mod=*/(short)0, acc, /*reuse_a=*/false, /*reuse_b=*/false);
    }
    // C/D: VGPR r -> row rowBase + r + 8*half, col colBase + l
    #pragma unroll
    for (int r = 0; r < 8; ++r) {
        int row = rowBase + r + 8 * half;
        if (row < Nn) Hout[(size_t)row * HC + colBase + l] = acc[r];
    }
}

// ---------------- per-node attention coefficients ----------------
__global__ void attn_coef(const float* __restrict__ Hmat,
                          const float* __restrict__ att_src,
                          const float* __restrict__ att_dst,
                          float* __restrict__ a_src, float* __restrict__ a_dst, int Nn) {
    int n = blockIdx.x * blockDim.x + threadIdx.x;
    if (n >= Nn) return;
    const float* hp = Hmat + (size_t)n * HC;
    #pragma unroll
    for (int hd = 0; hd < HEADS; ++hd) {
        float s = 0.0f, d = 0.0f;
        #pragma unroll 8
        for (int c = 0; c < CH; ++c) {
            float v = hp[hd * CH + c];
            s = fmaf(v, att_src[hd * CH + c], s);
            d = fmaf(v, att_dst[hd * CH + c], d);
        }
        a_src[n * HEADS + hd] = s;
        a_dst[n * HEADS + hd] = d;
    }
}

// ---- float <-> order-preserving unsigned (for atomic segment max) ----
__device__ __forceinline__ unsigned f2ord(float f) {
    unsigned u = __float_as_uint(f);
    return (u & 0x80000000u) ? ~u : (u | 0x80000000u);
}
__device__ __forceinline__ float ord2f(unsigned u) {
    unsigned b = (u & 0x80000000u) ? (u & 0x7FFFFFFFu) : ~u;
    return __uint_as_float(b);
}

// ---------------- edge pass A: leaky-relu logits + segment max ----------------
__global__ void edge_logits(const int* __restrict__ ei, int E_, int Nn,
                            const float* __restrict__ a_src, const float* __restrict__ a_dst,
                            float* __restrict__ ebuf, unsigned* __restrict__ m_u) {
    int i = blockIdx.x * blockDim.x + threadIdx.x;
    int Et = E_ + Nn;
    if (i >= Et) return;
    int s, d;
    if (i < E_) { s = ei[i]; d = ei[E_ + i]; } else { s = d = i - E_; }
    #pragma unroll
    for (int hd = 0; hd < HEADS; ++hd) {
        float e = a_src[s * HEADS + hd] + a_dst[d * HEADS + hd];
        e = (e > 0.0f) ? e : NEG_SLOPE * e;
        ebuf[(size_t)i * HEADS + hd] = e;
        atomicMax(&m_u[d * HEADS + hd], f2ord(e));
    }
}

// ---------------- edge pass B: exp(e - m) + segment sum ----------------
__global__ void edge_exp(const int* __restrict__ ei, int E_, int Nn,
                         const unsigned* __restrict__ m_u,
                         float* __restrict__ ebuf, float* __restrict__ den) {
    int i = blockIdx.x * blockDim.x + threadIdx.x;
    int Et = E_ + Nn;
    if (i >= Et) return;
    int d = (i < E_) ? ei[E_ + i] : (i - E_);
    #pragma unroll
    for (int hd = 0; hd < HEADS; ++hd) {
        float m = ord2f(m_u[d * HEADS + hd]);
        float ex = expf(ebuf[(size_t)i * HEADS + hd] - m);
        ebuf[(size_t)i * HEADS + hd] = ex;
        atomicAdd(&den[d * HEADS + hd], ex);
    }
}

// ---------------- edge pass C: weighted scatter, one wave per edge ----------------
__global__ __launch_bounds__(256) void edge_agg(const int* __restrict__ ei, int E_, int Nn,
                                                const float* __restrict__ ebuf,
                                                const float* __restrict__ den,
                                                const float* __restrict__ Hmat,
                                                float* __restrict__ agg) {
    int wid  = (blockIdx.x * blockDim.x + threadIdx.x) >> 5;
    int lane = threadIdx.x & 31;
    int Et = E_ + Nn;
    if (wid >= Et) return;
    int s, d;
    if (wid < E_) { s = ei[wid]; d = ei[E_ + wid]; } else { s = d = wid - E_; }
    int c0 = lane * 4;                 // 4 channels per lane, same head per lane
    int hd = c0 >> 6;
    float alpha = ebuf[(size_t)wid * HEADS + hd] / (den[d * HEADS + hd] + 1e-16f);
    const float4 hv = *(const float4*)(Hmat + (size_t)s * HC + c0);
    float* ap = agg + (size_t)d * HC + c0;
    atomicAdd(ap + 0, hv.x * alpha);
    atomicAdd(ap + 1, hv.y * alpha);
    atomicAdd(ap + 2, hv.z * alpha);
    atomicAdd(ap + 3, hv.w * alpha);
}

// ---------------- tail: relu -> fc -> log_softmax ----------------
__global__ __launch_bounds__(256) void head_fc(const float* __restrict__ agg,
                                               const float* __restrict__ bias,
                                               const float* __restrict__ Wfc,
                                               const float* __restrict__ bfc,
                                               float* __restrict__ out, int Nn) {
    __shared__ float sW[HC * NC];
    __shared__ float sBias[HC];
    __shared__ float sb[NC];
    for (int i = threadIdx.x; i < HC * NC; i += blockDim.x) sW[i] = Wfc[i];
    for (int i = threadIdx.x; i < HC; i += blockDim.x) sBias[i] = bias[i];
    if (threadIdx.x < NC) sb[threadIdx.x] = bfc[threadIdx.x];
    __syncthreads();

    for (int n = blockIdx.x * blockDim.x + threadIdx.x; n < Nn;
         n += gridDim.x * blockDim.x) {
        float acc[NC];
        #pragma unroll
        for (int j = 0; j < NC; ++j) acc[j] = sb[j];
        const float* ap = agg + (size_t)n * HC;
        for (int k = 0; k < HC; ++k) {
            float v = ap[k] + sBias[k];
            v = v > 0.0f ? v : 0.0f;
            #pragma unroll
            for (int j = 0; j < NC; ++j) acc[j] = fmaf(v, sW[k * NC + j], acc[j]);
        }
        float mx = acc[0];
        #pragma unroll
        for (int j = 1; j < NC; ++j) mx = fmaxf(mx, acc[j]);
        float sum = 0.0f;
        #pragma unroll
        for (int j = 0; j < NC; ++j) sum += expf(acc[j] - mx);
        float lse = mx + logf(sum);
        #pragma unroll
        for (int j = 0; j < NC; ++j) out[(size_t)n * NC + j] = acc[j] - lse;
    }
}

extern "C" void kernel_launch(void* const* d_in, const int* in_sizes, int n_in,
                              void* d_out, int out_size, void* d_ws, size_t ws_size,
                              hipStream_t stream) {
    const float* x       = (const float*)d_in[0];
    const int*   ei      = (const int*)  d_in[1];   // [2,E] int32
    const float* W       = (const float*)d_in[2];
    const float* att_src = (const float*)d_in[3];
    const float* att_dst = (const float*)d_in[4];
    const float* bias    = (const float*)d_in[5];
    const float* Wfc     = (const float*)d_in[6];
    const float* bfc     = (const float*)d_in[7];
    float* out = (float*)d_out;

    const int Nn = in_sizes[0] / F_IN;
    const int E_ = in_sizes[1] / 2;
    const int Et = E_ + Nn;

    // workspace layout
    float*    ws    = (float*)d_ws;
    float*    h     = ws;                                  // Nn*HC
    float*    agg   = h     + (size_t)Nn * HC;             // Nn*HC
    float*    a_src = agg   + (size_t)Nn * HC;             // Nn*HEADS
    float*    a_dst = a_src + (size_t)Nn * HEADS;          // Nn*HEADS
    float*    den   = a_dst + (size_t)Nn * HEADS;          // Nn*HEADS
    unsigned* m_u   = (unsigned*)(den + (size_t)Nn * HEADS); // Nn*HEADS
    float*    ebuf  = (float*)(m_u + (size_t)Nn * HEADS);  // Et*HEADS

    const int initN = Nn * HC;
    init_ws   <<<(initN + 255) / 256, 256, 0, stream>>>(agg, m_u, den, Nn);
    gemm_wmma <<<(Nn + 15) / 16,      256, 0, stream>>>(x, W, h, Nn);
    attn_coef <<<(Nn + 255) / 256,    256, 0, stream>>>(h, att_src, att_dst, a_src, a_dst, Nn);
    edge_logits<<<(Et + 255) / 256,   256, 0, stream>>>(ei, E_, Nn, a_src, a_dst, ebuf, m_u);
    edge_exp  <<<(Et + 255) / 256,    256, 0, stream>>>(ei, E_, Nn, m_u, ebuf, den);
    edge_agg  <<<(Et * 32 + 255)/256, 256, 0, stream>>>(ei, E_, Nn, ebuf, den, h, agg);
    head_fc   <<<512,                 256, 0, stream>>>(agg, bias, Wfc, bfc, out, Nn);
}